// AdderNet_22806276341978
// MI455X (gfx1250) — compile-verified
//
#include <hip/hip_runtime.h>
#include <hip/hip_bf16.h>

typedef __attribute__((ext_vector_type(2))) float v2f;
typedef __attribute__((ext_vector_type(8))) float v8f;

#define BN_EPS 1e-5f
#define CPB 4   // couts per block in adder2d (all layer Couts divisible by 4)

// ---------------------------------------------------------------------------
// Stem conv: 3x3, Cin=3, Cout=16, 32x32, SAME, stride 1, via WMMA f32 16x16x4.
// Implicit GEMM: M = pixel (16 consecutive pixels of one image row half),
// N = cout, K = ci*9+kh*3+kw (27, padded to 28).
// One wave per 16-pixel tile; 4 waves per block; 2048 tiles total.
// ---------------------------------------------------------------------------
__global__ void stem_conv_wmma_kernel(const float* __restrict__ x,
                                      const float* __restrict__ w,
                                      float* __restrict__ out) {
  int wave = threadIdx.x >> 5;
  int lane = threadIdx.x & 31;
  int tile = blockIdx.x * 4 + wave;          // 0..2047
  int pixbase = tile * 16;                   // aligned to 16 -> same image, same row
  int b  = pixbase >> 10;                    // /1024
  int rm = pixbase & 1023;
  int y  = rm >> 5;                          // row
  int xb = rm & 31;                          // 0 or 16
  int m  = lane & 15;                        // A-row (pixel within tile)
  int kb = (lane >> 4) << 1;                 // K sub-offset {0,2}
  int n  = lane & 15;                        // B-col (cout)
  int px = xb + m;                           // pixel x coordinate for A row m

  v8f acc = {0.f, 0.f, 0.f, 0.f, 0.f, 0.f, 0.f, 0.f};

  for (int step = 0; step < 7; ++step) {     // K = 28 (27 real + 1 zero pad)
    v2f a, bf;
#pragma unroll
    for (int j = 0; j < 2; ++j) {
      int k = step * 4 + kb + j;
      float av = 0.f, bv = 0.f;
      if (k < 27) {
        int ci = k / 9, r = k % 9, kh = r / 3, kw = r % 3;
        int yy = y + kh - 1, xx = px + kw - 1;
        if (yy >= 0 && yy < 32 && xx >= 0 && xx < 32)
          av = x[((b * 3 + ci) * 32 + yy) * 32 + xx];
        bv = w[n * 27 + k];                  // w: [16][3][3][3] flat, same k order
      }
      a[j] = av;
      bf[j] = bv;
    }
    acc = __builtin_amdgcn_wmma_f32_16x16x4_f32(false, a, false, bf,
                                                (short)0, acc, false, false);
  }

  int Mbase = (lane >> 4) << 3;
#pragma unroll
  for (int r8 = 0; r8 < 8; ++r8) {
    int M = r8 + Mbase;                      // pixel index within tile
    out[((b * 16 + n) * 32 + y) * 32 + (xb + M)] = acc[r8];
  }
}

// ---------------------------------------------------------------------------
// AdderNet pseudo-conv: out[b,co,yo,xo] = -sum_{ci,kh,kw} |patch - w|
// Each block computes CPB=4 output channels for 256 pixels of one image:
// one global load of the input value feeds 4 sub/abs/add chains (1 load : 8
// VALU ops), weights for the 4 couts staged in LDS (<= 4*576 floats = 9 KB).
// grid: (ceil(HoWo/256), Cout/CPB, B), block 256 (8 wave32).
// ---------------------------------------------------------------------------
__global__ void adder2d_kernel(const float* __restrict__ in,
                               const float* __restrict__ w,
                               float* __restrict__ out,
                               int Cin, int Hin, int Win,
                               int Cout, int Hout, int Wout,
                               int ksz, int stride, int pad) {
  __shared__ float wl[CPB * 576];
  int ck = Cin * ksz * ksz;
  int cob = blockIdx.y * CPB;
  for (int i = threadIdx.x; i < CPB * ck; i += blockDim.x)
    wl[i] = w[(size_t)cob * ck + i];
  __syncthreads();

  int b = blockIdx.z;
  int p = blockIdx.x * blockDim.x + threadIdx.x;
  if (p >= Hout * Wout) return;
  int yo = p / Wout, xo = p - yo * Wout;
  int yi0 = yo * stride - pad;
  int xi0 = xo * stride - pad;

  const float* w0 = wl;
  const float* w1 = wl + ck;
  const float* w2 = wl + 2 * ck;
  const float* w3 = wl + 3 * ck;
  float a0 = 0.f, a1 = 0.f, a2 = 0.f, a3 = 0.f;
  int wi = 0;
  for (int ci = 0; ci < Cin; ++ci) {
    const float* base = in + ((size_t)(b * Cin + ci) * Hin) * Win;
    for (int kh = 0; kh < ksz; ++kh) {
      int yy = yi0 + kh;
      bool yok = (yy >= 0) && (yy < Hin);
#pragma unroll 3
      for (int kw = 0; kw < ksz; ++kw) {
        int xx = xi0 + kw;
        float xv = (yok && xx >= 0 && xx < Win) ? base[yy * Win + xx] : 0.f;
        a0 -= fabsf(xv - w0[wi]);            // zero-padded patches contribute |0-w|
        a1 -= fabsf(xv - w1[wi]);
        a2 -= fabsf(xv - w2[wi]);
        a3 -= fabsf(xv - w3[wi]);
        ++wi;
      }
    }
  }
  size_t o = ((size_t)(b * Cout + cob) * Hout + yo) * Wout + xo;
  size_t cstride = (size_t)Hout * Wout;
  out[o]               = a0;
  out[o + cstride]     = a1;
  out[o + 2 * cstride] = a2;
  out[o + 3 * cstride] = a3;
}

// ---------------------------------------------------------------------------
// BN stats: one block per channel; params[2c] = mean, params[2c+1] = rsqrt(var+eps)
// ---------------------------------------------------------------------------
__global__ void bn_stats_kernel(const float* __restrict__ x,
                                float* __restrict__ params,
                                int B, int C, int HW) {
  __shared__ float ssum[256], ssq[256];
  int c = blockIdx.x;
  int tid = threadIdx.x;
  int N = B * HW;
  float s = 0.f, q = 0.f;
  for (int i = tid; i < N; i += 256) {
    int b = i / HW, r = i - b * HW;
    float v = x[(size_t)(b * C + c) * HW + r];
    s += v;
    q += v * v;
  }
  ssum[tid] = s; ssq[tid] = q;
  __syncthreads();
  for (int off = 128; off > 0; off >>= 1) {
    if (tid < off) { ssum[tid] += ssum[tid + off]; ssq[tid] += ssq[tid + off]; }
    __syncthreads();
  }
  if (tid == 0) {
    float m = ssum[0] / (float)N;
    float var = ssq[0] / (float)N - m * m;
    params[2 * c] = m;
    params[2 * c + 1] = rsqrtf(var + BN_EPS);
  }
}

__global__ void bn_apply_kernel(const float* __restrict__ x,
                                const float* __restrict__ params,
                                float* __restrict__ y,
                                int C, int HW, int total, int relu) {
  int i = blockIdx.x * blockDim.x + threadIdx.x;
  if (i >= total) return;
  int c = (i / HW) % C;
  float v = (x[i] - params[2 * c]) * params[2 * c + 1];
  if (relu) v = fmaxf(v, 0.f);
  y[i] = v;
}

// y = relu(bn(a) + identity)  (identity already finalized; y may alias id)
__global__ void bn_add_relu_kernel(const float* __restrict__ a,
                                   const float* __restrict__ params,
                                   const float* __restrict__ id,
                                   float* __restrict__ y,
                                   int C, int HW, int total) {
  int i = blockIdx.x * blockDim.x + threadIdx.x;
  if (i >= total) return;
  int c = (i / HW) % C;
  float v = (a[i] - params[2 * c]) * params[2 * c + 1] + id[i];
  y[i] = fmaxf(v, 0.f);
}

// global average over 8x8 map: one thread per (b, c)
__global__ void avgpool_kernel(const float* __restrict__ x,
                               float* __restrict__ out, int C, int HW, int nbc) {
  int idx = blockIdx.x * blockDim.x + threadIdx.x;
  if (idx >= nbc) return;
  const float* p = x + (size_t)idx * HW;
  float s = 0.f;
  for (int i = 0; i < HW; ++i) s += p[i];
  out[idx] = s / (float)HW;
}

// ---------------------------------------------------------------------------
// FC classifier: logits[32,10] = pool[32,64] * fc_w[10,64]^T via WMMA f32.
// 2 waves (one per 16-row M-tile), K = 64 -> 16 WMMA steps, N padded 10->16.
// ---------------------------------------------------------------------------
__global__ void fc_wmma_kernel(const float* __restrict__ pool,
                               const float* __restrict__ fcw,
                               float* __restrict__ logits) {
  int lane = threadIdx.x & 31;
  int wv = threadIdx.x >> 5;                 // M tile: rows wv*16 .. wv*16+15
  int m = lane & 15;
  int kb = (lane >> 4) << 1;
  int n = lane & 15;

  v8f acc = {0.f, 0.f, 0.f, 0.f, 0.f, 0.f, 0.f, 0.f};
  for (int step = 0; step < 16; ++step) {
    int k0 = step * 4 + kb;
    v2f a, bf;
    a[0] = pool[(wv * 16 + m) * 64 + k0];
    a[1] = pool[(wv * 16 + m) * 64 + k0 + 1];
    bf[0] = (n < 10) ? fcw[n * 64 + k0] : 0.f;
    bf[1] = (n < 10) ? fcw[n * 64 + k0 + 1] : 0.f;
    acc = __builtin_amdgcn_wmma_f32_16x16x4_f32(false, a, false, bf,
                                                (short)0, acc, false, false);
  }
  int Mbase = (lane >> 4) << 3;
#pragma unroll
  for (int r8 = 0; r8 < 8; ++r8) {
    int M = r8 + Mbase;
    if (n < 10) logits[(wv * 16 + M) * 10 + n] = acc[r8];
  }
}

// final BN over [32,10] logits (mean/var per class over batch), write d_out
__global__ void final_bn_kernel(const float* __restrict__ logits,
                                float* __restrict__ out) {
  int c = threadIdx.x;
  if (c >= 10) return;
  float s = 0.f, q = 0.f;
  for (int b = 0; b < 32; ++b) {
    float v = logits[b * 10 + c];
    s += v; q += v * v;
  }
  float m = s * (1.f / 32.f);
  float var = q * (1.f / 32.f) - m * m;
  float r = rsqrtf(var + BN_EPS);
  for (int b = 0; b < 32; ++b)
    out[b * 10 + c] = (logits[b * 10 + c] - m) * r;
}

// ---------------------------------------------------------------------------
extern "C" void kernel_launch(void* const* d_in, const int* in_sizes, int n_in,
                              void* d_out, int out_size, void* d_ws, size_t ws_size,
                              hipStream_t stream) {
  const float* x       = (const float*)d_in[0];
  const float* conv1_w = (const float*)d_in[1];
  const float* l1_w    = (const float*)d_in[2];  // [6][16][16*9]
  const float* l2_w0   = (const float*)d_in[3];  // [32][16*9]
  const float* l2_ws   = (const float*)d_in[4];  // [5][32][32*9]
  const float* l2_down = (const float*)d_in[5];  // [32][16]
  const float* l3_w0   = (const float*)d_in[6];  // [64][32*9]
  const float* l3_ws   = (const float*)d_in[7];  // [5][64][64*9]
  const float* l3_down = (const float*)d_in[8];  // [64][32]
  const float* fc_w    = (const float*)d_in[9];  // [10][64]
  float* out = (float*)d_out;

  const int B = 32;
  const size_t MAXT = (size_t)32 * 16 * 32 * 32;   // 524288 floats, largest tensor
  float* ws = (float*)d_ws;
  float* buf0   = ws;                  // h (block input / output)
  float* buf1   = buf0 + MAXT;         // conv1 raw / normalized
  float* buf2   = buf1 + MAXT;         // conv2 raw
  float* buf3   = buf2 + MAXT;         // downsample path
  float* params = buf3 + MAXT;         // [2*64] per-channel (mean, rstd)
  float* pool   = params + 128;        // [32*64]
  float* logits = pool + 2048;         // [32*10]

  auto adder = [&](const float* in, const float* w, float* o,
                   int Cin, int Hin, int Cout, int Hout, int k, int stride, int pad) {
    dim3 g((Hout * Hout + 255) / 256, Cout / CPB, B);
    adder2d_kernel<<<g, 256, 0, stream>>>(in, w, o, Cin, Hin, Hin,
                                          Cout, Hout, Hout, k, stride, pad);
  };
  auto bn_stats = [&](const float* t, int C, int HW) {
    bn_stats_kernel<<<C, 256, 0, stream>>>(t, params, B, C, HW);
  };
  auto bn_apply = [&](const float* src, float* dst, int C, int HW, int relu) {
    int total = B * C * HW;
    bn_apply_kernel<<<(total + 255) / 256, 256, 0, stream>>>(src, params, dst,
                                                             C, HW, total, relu);
  };

  // One residual block: input h in buf0 [B,Cin,Hin,Hin] -> output in buf0.
  auto block = [&](const float* w1, const float* w2, const float* down,
                   int Cin, int Hin, int Cout, int stride) {
    int Hout = Hin / stride;
    int HWo = Hout * Hout;
    int total = B * Cout * HWo;
    // out1 = relu(bn(adder(h, w1, stride, pad=1)))
    adder(buf0, w1, buf1, Cin, Hin, Cout, Hout, 3, stride, 1);
    bn_stats(buf1, Cout, HWo);
    bn_apply(buf1, buf1, Cout, HWo, 1);
    // identity (finalize BEFORE conv2 stats so `params` isn't clobbered)
    const float* ident = buf0;
    if (down) {
      adder(buf0, down, buf3, Cin, Hin, Cout, Hout, 1, stride, 0);
      bn_stats(buf3, Cout, HWo);
      bn_apply(buf3, buf3, Cout, HWo, 0);
      ident = buf3;
    }
    // out2 = bn(adder(out1, w2, 1, pad=1)); h = relu(out2 + identity)
    adder(buf1, w2, buf2, Cout, Hout, Cout, Hout, 3, 1, 1);
    bn_stats(buf2, Cout, HWo);
    bn_add_relu_kernel<<<(total + 255) / 256, 256, 0, stream>>>(
        buf2, params, ident, buf0, Cout, HWo, total);
  };

  // ---- stem: conv3x3 (WMMA) + BN + relu -> buf0 ----
  stem_conv_wmma_kernel<<<512, 128, 0, stream>>>(x, conv1_w, buf1);
  bn_stats(buf1, 16, 1024);
  bn_apply(buf1, buf0, 16, 1024, 1);

  // ---- layer1: 3 blocks, 16ch @ 32x32 ----
  const size_t W16 = (size_t)16 * 16 * 9;
  for (int i = 0; i < 3; ++i)
    block(l1_w + (size_t)(2 * i) * W16, l1_w + (size_t)(2 * i + 1) * W16,
          nullptr, 16, 32, 16, 1);

  // ---- layer2: 32ch @ 16x16, first block stride 2 + 1x1 adder downsample ----
  const size_t W32 = (size_t)32 * 32 * 9;
  block(l2_w0, l2_ws + 0 * W32, l2_down, 16, 32, 32, 2);
  block(l2_ws + 1 * W32, l2_ws + 2 * W32, nullptr, 32, 16, 32, 1);
  block(l2_ws + 3 * W32, l2_ws + 4 * W32, nullptr, 32, 16, 32, 1);

  // ---- layer3: 64ch @ 8x8 ----
  const size_t W64 = (size_t)64 * 64 * 9;
  block(l3_w0, l3_ws + 0 * W64, l3_down, 32, 16, 64, 2);
  block(l3_ws + 1 * W64, l3_ws + 2 * W64, nullptr, 64, 8, 64, 1);
  block(l3_ws + 3 * W64, l3_ws + 4 * W64, nullptr, 64, 8, 64, 1);

  // ---- head: global avg pool, FC (WMMA), final BN ----
  avgpool_kernel<<<(2048 + 255) / 256, 256, 0, stream>>>(buf0, pool, 64, 64, B * 64);
  fc_wmma_kernel<<<1, 64, 0, stream>>>(pool, fc_w, logits);
  final_bn_kernel<<<1, 16, 0, stream>>>(logits, out);
}